// SpectralConv2d_83167746719988
// MI455X (gfx1250) — compile-verified
//
#include <hip/hip_runtime.h>

typedef float v2f __attribute__((ext_vector_type(2)));
typedef float v8f __attribute__((ext_vector_type(8)));

// Problem dims
#define BB   32
#define MD   128
#define ND   128
#define CH   64
#define NM   16
#define KX   32   // row modes: 0..15 and 112..127
#define KY   16   // col (rfft) modes: 0..15

// Workspace layout (float offsets)
#define A1_OFF   0u                       // [32][128]  fwd col-DFT rows (ky,ri)
#define A5_OFF   4096u                    // [128][32]  inv col-DFT (real out)
#define FROW_OFF 8192u                    // [32][128][2] fwd row twiddles (c,s)
#define TINV_OFF 16384u                   // [128][32][2] inv row twiddles (c,s)
#define Y_OFF    24576u                   // [B][M][32(ky,ri)][64] = 8388608
#define Z_OFF    8413184u                 // [B][KX][KY][CH][2]   = 2097152
#define C1_OFF   10510336u                // [B][KX][KY][CH][2]   = 2097152
#define ZF_OFF   12607488u                // [B][KX][KY][CH][2]   = 2097152
#define U_OFF    14704640u                // [B][M][32(ky,ri)][64]= 8388608

__device__ __forceinline__ v8f wmma4(v2f a, v2f b, v8f c) {
  // D = A(16x4 f32) * B(4x16 f32) + C, full fp32 on the matrix pipe
  return __builtin_amdgcn_wmma_f32_16x16x4_f32(false, a, false, b, (short)0, c,
                                               false, false);
}

__device__ __forceinline__ v2f frag_a(const float* A, int ld, int row, int k, int half) {
  v2f a;
  a.x = A[row * ld + k + 2 * half + 0];
  a.y = A[row * ld + k + 2 * half + 1];
  return a;
}
__device__ __forceinline__ v2f frag_b(const float* Bm, int ld, int k, int col, int half) {
  v2f b;
  b.x = Bm[(k + 2 * half + 0) * ld + col];
  b.y = Bm[(k + 2 * half + 1) * ld + col];
  return b;
}

// ---- CDNA5 async global->LDS copy (ASYNCcnt path, no VGPR staging) --------
// Generic LDS pointer low 32 bits == wave-relative LDS byte offset.
__device__ __forceinline__ unsigned lds_byte_off(const void* p) {
  return (unsigned)(size_t)p;
}
__device__ __forceinline__ void async_b128(unsigned lds_byte, const void* g) {
  asm volatile("global_load_async_to_lds_b128 %0, %1, off"
               :: "v"(lds_byte), "v"(g)
               : "memory");
}
__device__ __forceinline__ void wait_async0() {
  asm volatile("s_wait_asynccnt 0" ::: "memory");
}

// ---------------------------------------------------------------------------
// Twiddle / coefficient tables. Ortho 1/sqrt(M*N)=1/128 folded per direction.
// Phases use exact integer mod-128 reduction before sin/cos.
// ---------------------------------------------------------------------------
__global__ void fno_init_tables(float* __restrict__ ws) {
  const float W = 6.283185307179586f / 128.0f;
  const float S = 1.0f / 128.0f;
  int tid = blockIdx.x * blockDim.x + threadIdx.x;
  if (tid < 4096) {                      // A1[(2ky+ri)][n]
    int row = tid >> 7, n = tid & 127;
    int ky = row >> 1, ri = row & 1;
    float th = (float)((n * ky) & 127) * W;
    ws[A1_OFF + tid] = (ri == 0 ? __cosf(th) : -__sinf(th)) * S;
  } else if (tid < 8192) {               // A5[n][2ky+ri]
    int t = tid - 4096;
    int n = t >> 5, kk = t & 31;
    int ky = kk >> 1, ri = kk & 1;
    float th = (float)((n * ky) & 127) * W;
    float v;
    if (ky == 0) v = (ri == 0) ? S : 0.0f;
    else         v = (ri == 0) ? 2.0f * S * __cosf(th) : -2.0f * S * __sinf(th);
    ws[A5_OFF + t] = v;
  } else if (tid < 16384) {              // FROW[x][m][{c,s}], e^{-i th}
    int t = tid - 8192;
    int x = t >> 8, rem = t & 255;
    int m = rem >> 1, ri = rem & 1;
    int kxv = (x < 16) ? x : x + 96;
    float th = (float)((m * kxv) & 127) * W;
    ws[FROW_OFF + t] = (ri == 0) ? __cosf(th) : __sinf(th);
  } else if (tid < 24576) {              // TINV[m][x][{c,s}], e^{+i th}
    int t = tid - 16384;
    int m = t >> 6, rem = t & 63;
    int x = rem >> 1, ri = rem & 1;
    int kxv = (x < 16) ? x : x + 96;
    float th = (float)((m * kxv) & 127) * W;
    ws[TINV_OFF + t] = (ri == 0) ? __cosf(th) : __sinf(th);
  }
}

// ---------------------------------------------------------------------------
// Stage 1: per (b,m): Y[32,64] = A1[32x128] * x[b,m][128n x 64i]  (WMMA f32)
// ---------------------------------------------------------------------------
__global__ __launch_bounds__(128) void fno_stage1(const float* __restrict__ x,
                                                  const float* __restrict__ ws,
                                                  float* __restrict__ Y) {
  __shared__ float sA[32 * 128];
  __shared__ float sX[128 * 64];
  int slice = blockIdx.x;                          // b*128 + m
  const float* xs = x + (size_t)slice * (ND * CH);
  unsigned sA_b = lds_byte_off(sA), sX_b = lds_byte_off(sX);
  for (unsigned i = threadIdx.x * 4; i < 4096; i += 128 * 4)
    async_b128(sA_b + i * 4, ws + A1_OFF + i);
  for (unsigned i = threadIdx.x * 4; i < 8192; i += 128 * 4)
    async_b128(sX_b + i * 4, xs + i);
  wait_async0();
  __syncthreads();
  int lane = threadIdx.x & 31, wave = threadIdx.x >> 5;
  int half = lane >> 4, lr = lane & 15;
  float* yo = Y + (size_t)slice * (32 * CH);
  int ct = wave;                                    // 4 col tiles
  for (int rt = 0; rt < 2; ++rt) {                  // 2 row tiles
    v8f acc = {0, 0, 0, 0, 0, 0, 0, 0};
    for (int k = 0; k < 128; k += 4) {
      v2f a = frag_a(sA, 128, rt * 16 + lr, k, half);
      v2f b = frag_b(sX, 64, k, ct * 16 + lr, half);
      acc = wmma4(a, b, acc);
    }
    for (int r = 0; r < 8; ++r)
      yo[(rt * 16 + half * 8 + r) * CH + ct * 16 + lr] = acc[r];
  }
}

// ---------------------------------------------------------------------------
// Stage 2: forward row DFT (complex), VALU.
// Z[b][x][y][i] = sum_m Y[b][m][(y,ri)][i] * e^{-i th(m,x)}
// ---------------------------------------------------------------------------
__global__ __launch_bounds__(256) void fno_stage2(const float* __restrict__ ws,
                                                  float* __restrict__ Z) {
  int tid = blockIdx.x * blockDim.x + threadIdx.x;  // (((b*32+x)*16+y)*64+i)
  int i = tid & 63, y = (tid >> 6) & 15, x = (tid >> 10) & 31, b = tid >> 15;
  const float* Yb = ws + Y_OFF + (size_t)b * (MD * 32 * CH);
  const float* tw = ws + FROW_OFF + x * 256;
  float zr = 0.f, zi = 0.f;
  for (int m = 0; m < 128; ++m) {
    if ((m & 15) == 0 && m + 16 < 128)
      __builtin_prefetch(&Yb[((m + 16) * 32 + y * 2) * CH + i], 0, 1);
    float yr = Yb[(m * 32 + y * 2 + 0) * CH + i];
    float yi = Yb[(m * 32 + y * 2 + 1) * CH + i];
    float c = tw[m * 2 + 0], s = tw[m * 2 + 1];
    zr += yr * c + yi * s;                          // (yr+iyi)(c-is)
    zi += yi * c - yr * s;
  }
  Z[(size_t)tid * 2 + 0] = zr;
  Z[(size_t)tid * 2 + 1] = zi;
}

// ---------------------------------------------------------------------------
// Mix 1: c[b,o,x,y] = sum_i (w0|w2)[i,o,x] * z[b,i,x,y]   (complex)
// ---------------------------------------------------------------------------
__global__ __launch_bounds__(256) void fno_mix1(const float* __restrict__ ws,
                                                const float* __restrict__ w0,
                                                const float* __restrict__ w2,
                                                float* __restrict__ C1) {
  int tid = blockIdx.x * blockDim.x + threadIdx.x;  // (((b*32+x)*16+y)*64+o)
  int o = tid & 63, y = (tid >> 6) & 15, x = (tid >> 10) & 31, b = tid >> 15;
  const float* Zs = ws + Z_OFF + ((size_t)(((b * 32 + x) * 16 + y) * 64)) * 2;
  const float* w = (x < 16) ? w0 : w2;
  int xp = x & 15;
  float cr = 0.f, ci = 0.f;
  for (int i = 0; i < 64; ++i) {
    float zr = Zs[i * 2 + 0], zi = Zs[i * 2 + 1];
    const float* wp = w + (size_t)((i * 64 + o) * 16 + xp) * 2;
    float wr = wp[0], wi = wp[1];
    cr += wr * zr - wi * zi;
    ci += wr * zi + wi * zr;
  }
  C1[(size_t)tid * 2 + 0] = cr;
  C1[(size_t)tid * 2 + 1] = ci;
}

// ---------------------------------------------------------------------------
// Mix 2: zf[b,o,x,y] = sum_i c[b,i,x,y] * (w1|w3)[i,o,y]   (complex)
// ---------------------------------------------------------------------------
__global__ __launch_bounds__(256) void fno_mix2(const float* __restrict__ ws,
                                                const float* __restrict__ w1,
                                                const float* __restrict__ w3,
                                                float* __restrict__ ZF) {
  int tid = blockIdx.x * blockDim.x + threadIdx.x;
  int o = tid & 63, y = (tid >> 6) & 15, x = (tid >> 10) & 31, b = tid >> 15;
  const float* Cs = ws + C1_OFF + ((size_t)(((b * 32 + x) * 16 + y) * 64)) * 2;
  const float* w = (x < 16) ? w1 : w3;
  float cr = 0.f, ci = 0.f;
  for (int i = 0; i < 64; ++i) {
    float zr = Cs[i * 2 + 0], zi = Cs[i * 2 + 1];
    const float* wp = w + (size_t)((i * 64 + o) * 16 + y) * 2;
    float wr = wp[0], wi = wp[1];
    cr += wr * zr - wi * zi;
    ci += wr * zi + wi * zr;
  }
  ZF[(size_t)tid * 2 + 0] = cr;
  ZF[(size_t)tid * 2 + 1] = ci;
}

// ---------------------------------------------------------------------------
// Stage 4: inverse row transform (complex), VALU.
// U[b][m][(ky,ri)][o] = sum_x zf[b][x][ky][o] * e^{+i th(m,x)}
// ---------------------------------------------------------------------------
__global__ __launch_bounds__(256) void fno_stage4(const float* __restrict__ ws,
                                                  float* __restrict__ U) {
  int tid = blockIdx.x * blockDim.x + threadIdx.x;  // (((b*128+m)*16+ky)*64+o)
  int o = tid & 63, ky = (tid >> 6) & 15, m = (tid >> 10) & 127, b = tid >> 17;
  const float* tw = ws + TINV_OFF + m * 64;
  const float* ZFb = ws + ZF_OFF + (size_t)b * (KX * KY * CH * 2);
  float ur = 0.f, ui = 0.f;
  for (int x = 0; x < 32; ++x) {
    size_t off = ((size_t)(x * 16 + ky) * 64 + o) * 2;
    float zr = ZFb[off + 0];
    float zi = ZFb[off + 1];
    float c = tw[x * 2 + 0], s = tw[x * 2 + 1];
    ur += zr * c - zi * s;                          // (zr+izi)(c+is)
    ui += zr * s + zi * c;
  }
  float* Ub = U + (size_t)(b * 128 + m) * (32 * CH);
  Ub[(ky * 2 + 0) * CH + o] = ur;
  Ub[(ky * 2 + 1) * CH + o] = ui;
}

// ---------------------------------------------------------------------------
// Stage 5 + dual MLP, fused per (b,m) workgroup, all WMMA f32.
//   XO[128n x 64o] = A5[128x32] * U[32x64]              (inverse col DFT)
//   H  = relu(XO * W1 + b1)   [128 x 128]
//   OUT = H * W2 + b2         [128 x 64]  -> d_out[mlp][b][m][n][o]
// LDS: XO 32K, H 64K, W 32K, bias 512B, A5 16K, U 8K  (~153 KB dynamic)
// All LDS staging uses GLOBAL_LOAD_ASYNC_TO_LDS_B128 (ASYNCcnt).
// ---------------------------------------------------------------------------
__global__ __launch_bounds__(256) void fno_stage5_mlp(
    const float* __restrict__ ws,
    const float* __restrict__ bw1, const float* __restrict__ bb1,
    const float* __restrict__ bw2, const float* __restrict__ bb2,
    const float* __restrict__ fw1, const float* __restrict__ fb1,
    const float* __restrict__ fw2, const float* __restrict__ fb2,
    float* __restrict__ out) {
  extern __shared__ float sm[];
  float* sXO = sm;              // 8192
  float* sH  = sm + 8192;       // 16384
  float* sW  = sm + 24576;      // 8192
  float* sB  = sm + 32768;      // 128
  float* sA5 = sm + 32896;      // 4096
  float* sU  = sm + 36992;      // 2048  -> total 39040 floats

  int slice = blockIdx.x;                  // b*128 + m
  int b = slice >> 7, m = slice & 127;
  int lane = threadIdx.x & 31, wave = threadIdx.x >> 5;
  int half = lane >> 4, lr = lane & 15;
  unsigned sA5_b = lds_byte_off(sA5), sU_b = lds_byte_off(sU);
  unsigned sW_b = lds_byte_off(sW);

  const float* Us = ws + U_OFF + (size_t)slice * (32 * CH);
  for (unsigned i = threadIdx.x * 4; i < 4096; i += 256 * 4)
    async_b128(sA5_b + i * 4, ws + A5_OFF + i);
  for (unsigned i = threadIdx.x * 4; i < 2048; i += 256 * 4)
    async_b128(sU_b + i * 4, Us + i);
  wait_async0();
  __syncthreads();

  // ---- stage 5: XO = A5 * U ----
  {
    int rt = wave;                         // 8 row tiles of 16
    for (int ct = 0; ct < 4; ++ct) {       // 64 cols
      v8f acc = {0, 0, 0, 0, 0, 0, 0, 0};
      for (int k = 0; k < 32; k += 4) {
        v2f a = frag_a(sA5, 32, rt * 16 + lr, k, half);
        v2f bfr = frag_b(sU, 64, k, ct * 16 + lr, half);
        acc = wmma4(a, bfr, acc);
      }
      for (int r = 0; r < 8; ++r)
        sXO[(rt * 16 + half * 8 + r) * 64 + ct * 16 + lr] = acc[r];
    }
  }
  __syncthreads();

  for (int mlp = 0; mlp < 2; ++mlp) {
    const float* W1 = mlp ? fw1 : bw1;
    const float* B1 = mlp ? fb1 : bb1;
    const float* W2 = mlp ? fw2 : bw2;
    const float* B2 = mlp ? fb2 : bb2;

    for (unsigned i = threadIdx.x * 4; i < 8192; i += 256 * 4)
      async_b128(sW_b + i * 4, W1 + i);
    for (int i = threadIdx.x; i < 128; i += 256) sB[i] = B1[i];
    wait_async0();
    __syncthreads();

    // layer 1: H = relu(XO[128x64] * W1[64x128] + b1)
    {
      int rt = wave;
      for (int ct = 0; ct < 8; ++ct) {
        v8f acc = {0, 0, 0, 0, 0, 0, 0, 0};
        for (int k = 0; k < 64; k += 4) {
          v2f a = frag_a(sXO, 64, rt * 16 + lr, k, half);
          v2f bfr = frag_b(sW, 128, k, ct * 16 + lr, half);
          acc = wmma4(a, bfr, acc);
        }
        float bias = sB[ct * 16 + lr];
        for (int r = 0; r < 8; ++r) {
          float v = acc[r] + bias;
          sH[(rt * 16 + half * 8 + r) * 128 + ct * 16 + lr] = v > 0.f ? v : 0.f;
        }
      }
    }
    __syncthreads();

    for (unsigned i = threadIdx.x * 4; i < 8192; i += 256 * 4)
      async_b128(sW_b + i * 4, W2 + i);
    for (int i = threadIdx.x; i < 64; i += 256) sB[i] = B2[i];
    wait_async0();
    __syncthreads();

    // layer 2: OUT = H[128x128] * W2[128x64] + b2 -> global
    float* outp = out + (((size_t)mlp * BB + b) * MD + m) * (ND * CH);
    {
      int rt = wave;
      for (int ct = 0; ct < 4; ++ct) {
        v8f acc = {0, 0, 0, 0, 0, 0, 0, 0};
        for (int k = 0; k < 128; k += 4) {
          v2f a = frag_a(sH, 128, rt * 16 + lr, k, half);
          v2f bfr = frag_b(sW, 64, k, ct * 16 + lr, half);
          acc = wmma4(a, bfr, acc);
        }
        float bias = sB[ct * 16 + lr];
        for (int r = 0; r < 8; ++r)
          outp[(rt * 16 + half * 8 + r) * 64 + ct * 16 + lr] = acc[r] + bias;
      }
    }
    __syncthreads();
  }
}

// ---------------------------------------------------------------------------
extern "C" void kernel_launch(void* const* d_in, const int* in_sizes, int n_in,
                              void* d_out, int out_size, void* d_ws, size_t ws_size,
                              hipStream_t stream) {
  (void)in_sizes; (void)n_in; (void)out_size; (void)ws_size;
  const float* x   = (const float*)d_in[0];
  const float* w0  = (const float*)d_in[1];
  const float* w1  = (const float*)d_in[2];
  const float* w2  = (const float*)d_in[3];
  const float* w3  = (const float*)d_in[4];
  const float* bw1 = (const float*)d_in[5];
  const float* bb1 = (const float*)d_in[6];
  const float* bw2 = (const float*)d_in[7];
  const float* bb2 = (const float*)d_in[8];
  const float* fw1 = (const float*)d_in[9];
  const float* fb1 = (const float*)d_in[10];
  const float* fw2 = (const float*)d_in[11];
  const float* fb2 = (const float*)d_in[12];
  float* out = (float*)d_out;
  float* ws  = (float*)d_ws;

  fno_init_tables<<<96, 256, 0, stream>>>(ws);
  fno_stage1<<<BB * MD, 128, 0, stream>>>(x, ws, ws + Y_OFF);
  fno_stage2<<<4096, 256, 0, stream>>>(ws, ws + Z_OFF);
  fno_mix1<<<4096, 256, 0, stream>>>(ws, w0, w2, ws + C1_OFF);
  fno_mix2<<<4096, 256, 0, stream>>>(ws, w1, w3, ws + ZF_OFF);
  fno_stage4<<<16384, 256, 0, stream>>>(ws, ws + U_OFF);
  size_t smem = 39040 * sizeof(float);
  fno_stage5_mlp<<<BB * MD, 256, smem, stream>>>(ws, bw1, bb1, bw2, bb2,
                                                 fw1, fb1, fw2, fb2, out);
}